// DecodeAttention_60189671686564
// MI455X (gfx1250) — compile-verified
//
#include <hip/hip_runtime.h>
#include <math.h>

#define HID    4096
#define NH     32
#define NKV    8
#define NREP   4
#define DH     128
#define MAXSEQ 8192
#define NSPLIT 64
#define SPLIT  (MAXSEQ / NSPLIT)   // 128 keys per split
#define EPSF   1e-6f

typedef __attribute__((ext_vector_type(2))) float v2f;
typedef __attribute__((ext_vector_type(8))) float v8f;

// ---------------------------------------------------------------------------
// GEMV: y[r] = dot(W[r, 0:HID], x)   (W row-major, coalesced b128 streaming,
// x staged in LDS; one wave per output row, 8 rows per 256-thread block)
// ---------------------------------------------------------------------------
__global__ __launch_bounds__(256) void gemv_kernel(const float* __restrict__ W,
                                                   const float* __restrict__ x,
                                                   float* __restrict__ y,
                                                   int rows) {
  __shared__ float xs[HID];
  const int tid = threadIdx.x;
  for (int i = tid * 4; i < HID; i += 256 * 4)
    *(float4*)&xs[i] = *(const float4*)&x[i];
  __syncthreads();

  const int wave = tid >> 5, lane = tid & 31;
  const int row = blockIdx.x * 8 + wave;
  if (row >= rows) return;

  const float* wr = W + (size_t)row * HID;
  float acc = 0.f;
#pragma unroll 4
  for (int c = lane * 4; c < HID; c += 32 * 4) {
    float4 wv = *(const float4*)&wr[c];
    float4 xv = *(const float4*)&xs[c];
    acc += wv.x * xv.x + wv.y * xv.y + wv.z * xv.z + wv.w * xv.w;
  }
#pragma unroll
  for (int off = 16; off > 0; off >>= 1) acc += __shfl_down(acc, off, 32);
  if (lane == 0) y[row] = acc;
}

// ---------------------------------------------------------------------------
// Per-head RMSNorm + RoPE for q/k, plus k/v cache-row update at `position`.
// One wave per head: blocks 0..31 = q heads, 32..39 = k heads, 40..47 = v heads.
// lane holds d = 4*lane .. 4*lane+3 ; RoPE partner (d +/- 64) sits at lane^16.
// ---------------------------------------------------------------------------
__global__ __launch_bounds__(32) void norm_rope_update_kernel(
    const float* __restrict__ qkv_raw,  // [NH*DH + NKV*DH + NKV*DH]
    const float* __restrict__ cosv, const float* __restrict__ sinv,
    const float* __restrict__ qn_w, const float* __restrict__ kn_w,
    const int* __restrict__ pos_ptr,
    float* __restrict__ q_final,        // [NH][DH]
    float* __restrict__ k_out,          // [NKV][MAXSEQ][DH]
    float* __restrict__ v_out) {        // [NKV][MAXSEQ][DH]
  const int b = blockIdx.x;
  const int lane = threadIdx.x;
  const int pos = *pos_ptr;

  const float* src;
  if (b < NH)            src = qkv_raw + b * DH;
  else if (b < NH + NKV) src = qkv_raw + NH * DH + (b - NH) * DH;
  else                   src = qkv_raw + (NH + NKV) * DH + (b - NH - NKV) * DH;

  float4 xv = *(const float4*)&src[lane * 4];

  if (b >= NH + NKV) {  // v head: plain copy into the cache row
    *(float4*)&v_out[((size_t)(b - NH - NKV) * MAXSEQ + pos) * DH + lane * 4] = xv;
    return;
  }

  // RMS norm over D=128
  float ss = xv.x * xv.x + xv.y * xv.y + xv.z * xv.z + xv.w * xv.w;
#pragma unroll
  for (int off = 16; off > 0; off >>= 1) ss += __shfl_xor(ss, off, 32);
  const float inv = rsqrtf(ss * (1.0f / (float)DH) + EPSF);
  const float* nw = (b < NH) ? qn_w : kn_w;

  float xn[4];
  xn[0] = xv.x * inv * nw[lane * 4 + 0];
  xn[1] = xv.y * inv * nw[lane * 4 + 1];
  xn[2] = xv.z * inv * nw[lane * 4 + 2];
  xn[3] = xv.w * inv * nw[lane * 4 + 3];

  // RoPE: out[d] = xn[d]*cos[d] + (d<64 ? -xn[d+64] : xn[d-64]) * sin[d]
  const float sgn = (lane < 16) ? -1.f : 1.f;
  float out[4];
#pragma unroll
  for (int j = 0; j < 4; ++j) {
    const int d = lane * 4 + j;
    const float part = __shfl_xor(xn[j], 16, 32);
    out[j] = xn[j] * cosv[d] + sgn * part * sinv[d];
  }
  float4 ov = make_float4(out[0], out[1], out[2], out[3]);
  if (b < NH) *(float4*)&q_final[b * DH + lane * 4] = ov;
  else *(float4*)&k_out[((size_t)(b - NH) * MAXSEQ + pos) * DH + lane * 4] = ov;
}

// ---------------------------------------------------------------------------
// Flash-decode partial: one wave per (kv-head, split of 128 keys).
// QK^T via V_WMMA_F32_16X16X4_F32, accumulated over d in 32 chunks of 4.
//   A (16x4): lane 0-15 holds K[key=lane][d0..d0+1], lane 16-31 d0+2..d0+3
//   B (4x16): column = q head (only 0..3 valid, rest zero)
//   C (16x16): VGPR r, lane n -> score[key = r + 8*(n>>4)][head = n&15]
// Online softmax per head (lanes n and n^16 cooperate), P*V in VALU with
// shfl broadcast of p values; acc distributed d = 4*lane+j across lanes.
// ---------------------------------------------------------------------------
__global__ __launch_bounds__(32) void attn_partial_kernel(
    const float* __restrict__ qf,        // [NH][DH]
    const float* __restrict__ kc,        // [NKV][MAXSEQ][DH] (updated cache)
    const float* __restrict__ vc,
    const int* __restrict__ pos_ptr,
    float* __restrict__ part_acc,        // [NKV][NSPLIT][NREP][DH]
    float* __restrict__ part_ml) {       // [NKV][NSPLIT][NREP][2]
  const int h = blockIdx.x;
  const int s = blockIdx.y;
  const int lane = threadIdx.x;
  const int pos = *pos_ptr;
  const int kbeg = s * SPLIT;
  const int kend = min(kbeg + SPLIT, pos + 1);
  const float scale = 0.08838834764831845f;  // 1/sqrt(128)

  const int col  = lane & 15;        // WMMA column / key-in-tile index
  const int koff = (lane >> 4) * 2;  // which 2 of the 4 k-slice this lane holds
  const int hi   = lane >> 4;

  // Stage q into B-operand register layout (64 VGPRs, reused for all tiles)
  const bool qv = (col < NREP);
  const float* qrow = qf + (h * NREP + (qv ? col : 0)) * DH;
  float bq0[32], bq1[32];
#pragma unroll
  for (int j = 0; j < 32; ++j) {
    bq0[j] = qv ? qrow[4 * j + koff]     : 0.f;
    bq1[j] = qv ? qrow[4 * j + koff + 1] : 0.f;
  }

  float mrun = -1e30f, lrun = 0.f;
  float acc[NREP][4];
#pragma unroll
  for (int r = 0; r < NREP; ++r)
#pragma unroll
    for (int j = 0; j < 4; ++j) acc[r][j] = 0.f;

  const float* kh = kc + (size_t)h * MAXSEQ * DH;
  const float* vh = vc + (size_t)h * MAXSEQ * DH;

  for (int t = 0; t < SPLIT / 16; ++t) {
    const int k0 = kbeg + t * 16;
    if (k0 >= kend) break;

    if (k0 + 16 < kend)  // prefetch next K tile row -> global_prefetch_b8
      __builtin_prefetch(&kh[(size_t)(k0 + 16 + col) * DH], 0, 1);

    const float* krp = kh + (size_t)(k0 + col) * DH + koff;
    v8f c = {};
#pragma unroll
    for (int j = 0; j < 32; ++j) {
      v2f a; a.x = krp[4 * j]; a.y = krp[4 * j + 1];
      v2f b; b.x = bq0[j];     b.y = bq1[j];
      c = __builtin_amdgcn_wmma_f32_16x16x4_f32(
          /*neg_a=*/false, a, /*neg_b=*/false, b,
          /*c_mod=*/(short)0, c, /*reuse_a=*/false, /*reuse_b=*/false);
    }

    // per-head online softmax over this 16-key tile
    float p[8];
    float smax = -1e30f;
#pragma unroll
    for (int r = 0; r < 8; ++r) {
      const int key = k0 + r + 8 * hi;
      const float sc = (key < kend) ? c[r] * scale : -1e30f;
      p[r] = sc;
      smax = fmaxf(smax, sc);
    }
    smax = fmaxf(smax, __shfl_xor(smax, 16, 32));
    const float mnew  = fmaxf(mrun, smax);
    const float alpha = __expf(mrun - mnew);
    float psum = 0.f;
#pragma unroll
    for (int r = 0; r < 8; ++r) {
      const int key = k0 + r + 8 * hi;
      const float e = (key < kend) ? __expf(p[r] - mnew) : 0.f;
      p[r] = e;
      psum += e;
    }
    lrun = lrun * alpha + psum;
    mrun = mnew;

#pragma unroll
    for (int r = 0; r < NREP; ++r) {
      const float ar = __shfl(alpha, r, 32);
#pragma unroll
      for (int j = 0; j < 4; ++j) acc[r][j] *= ar;
    }

    // P*V: lane owns d = 4*lane..4*lane+3
    for (int m = 0; m < 16; ++m) {
      const int key = k0 + m;
      if (key >= kend) break;
      const float4 vrow = *(const float4*)&vh[(size_t)key * DH + lane * 4];
#pragma unroll
      for (int r = 0; r < NREP; ++r) {
        const float pv = __shfl(p[m & 7], r + 16 * (m >> 3), 32);
        acc[r][0] += pv * vrow.x;
        acc[r][1] += pv * vrow.y;
        acc[r][2] += pv * vrow.z;
        acc[r][3] += pv * vrow.w;
      }
    }
  }

  const float lsum = lrun + __shfl_xor(lrun, 16, 32);
  const int base = (h * NSPLIT + s) * NREP;
  if (lane < NREP) {
    part_ml[(base + lane) * 2 + 0] = mrun;
    part_ml[(base + lane) * 2 + 1] = lsum;
  }
#pragma unroll
  for (int r = 0; r < NREP; ++r)
    *(float4*)&part_acc[(size_t)(base + r) * DH + lane * 4] =
        make_float4(acc[r][0], acc[r][1], acc[r][2], acc[r][3]);
}

// ---------------------------------------------------------------------------
// Combine NSPLIT partials per q head: renormalize by global max, divide by L.
// ---------------------------------------------------------------------------
__global__ __launch_bounds__(128) void attn_reduce_kernel(
    const float* __restrict__ part_acc, const float* __restrict__ part_ml,
    float* __restrict__ attn_out) {       // [NH][DH], head-major
  const int h = blockIdx.x, r = blockIdx.y, d = threadIdx.x;
  float M = -1e30f;
  for (int s = 0; s < NSPLIT; ++s)
    M = fmaxf(M, part_ml[((h * NSPLIT + s) * NREP + r) * 2]);
  float L = 0.f, o = 0.f;
  for (int s = 0; s < NSPLIT; ++s) {
    const int idx = (h * NSPLIT + s) * NREP + r;
    const float ms = part_ml[idx * 2 + 0];
    const float ls = part_ml[idx * 2 + 1];
    const float w = __expf(ms - M);
    L += ls * w;
    o += part_acc[(size_t)idx * DH + d] * w;
  }
  attn_out[(h * NREP + r) * DH + d] = o / L;
}

// ---------------------------------------------------------------------------
extern "C" void kernel_launch(void* const* d_in, const int* in_sizes, int n_in,
                              void* d_out, int out_size, void* d_ws, size_t ws_size,
                              hipStream_t stream) {
  (void)in_sizes; (void)n_in; (void)out_size; (void)ws_size;

  const float* hidden  = (const float*)d_in[0];
  const int*   pos     = (const int*)d_in[1];
  const float* cosv    = (const float*)d_in[2];
  const float* sinv    = (const float*)d_in[3];
  const float* k_cache = (const float*)d_in[4];
  const float* v_cache = (const float*)d_in[5];
  const float* q_w     = (const float*)d_in[6];
  const float* k_w     = (const float*)d_in[7];
  const float* v_w     = (const float*)d_in[8];
  const float* o_w     = (const float*)d_in[9];
  const float* qn_w    = (const float*)d_in[10];
  const float* kn_w    = (const float*)d_in[11];

  float* out   = (float*)d_out;                       // [4096]
  float* k_out = out + HID;                           // [NKV][MAXSEQ][DH]
  float* v_out = k_out + (size_t)NKV * MAXSEQ * DH;   // [NKV][MAXSEQ][DH]

  float* ws       = (float*)d_ws;
  float* qkv_raw  = ws;           // 6144
  float* q_final  = ws + 6144;    // 4096
  float* attn_out = ws + 10240;   // 4096
  float* part_ml  = ws + 14336;   // 8*64*4*2   = 4096
  float* part_acc = ws + 18432;   // 8*64*4*128 = 262144

  // Pass caches through to outputs (row at `position` overwritten below).
  const size_t cache_bytes = (size_t)NKV * MAXSEQ * DH * sizeof(float);
  hipMemcpyAsync(k_out, k_cache, cache_bytes, hipMemcpyDeviceToDevice, stream);
  hipMemcpyAsync(v_out, v_cache, cache_bytes, hipMemcpyDeviceToDevice, stream);

  // QKV projections (bandwidth-bound GEMVs over 100 MB of weights)
  gemv_kernel<<<512, 256, 0, stream>>>(q_w, hidden, qkv_raw, NH * DH);
  gemv_kernel<<<128, 256, 0, stream>>>(k_w, hidden, qkv_raw + NH * DH, NKV * DH);
  gemv_kernel<<<128, 256, 0, stream>>>(v_w, hidden, qkv_raw + (NH + NKV) * DH, NKV * DH);

  // RMSNorm + RoPE + cache-row update
  norm_rope_update_kernel<<<48, 32, 0, stream>>>(qkv_raw, cosv, sinv, qn_w, kn_w,
                                                 pos, q_final, k_out, v_out);

  // Flash-decode attention with WMMA QK^T
  attn_partial_kernel<<<dim3(NKV, NSPLIT), 32, 0, stream>>>(q_final, k_out, v_out,
                                                            pos, part_acc, part_ml);
  attn_reduce_kernel<<<dim3(NKV, NREP), 128, 0, stream>>>(part_acc, part_ml, attn_out);

  // Output projection
  gemv_kernel<<<512, 256, 0, stream>>>(o_w, attn_out, out, HID);
}